// NearestNeighborAttention_84507776516097
// MI455X (gfx1250) — compile-verified
//
#include <hip/hip_runtime.h>
#include <hip/hip_bf16.h>

// ---------------- problem constants (match reference) ----------------
constexpr int B  = 2;
constexpr int S  = 4096;
constexpr int F  = 256;
constexpr int H  = 8;
constexpr int D  = 32;    // head dim
constexpr int NK = 32;    // neighbors

typedef __attribute__((ext_vector_type(2))) float v2f;
typedef __attribute__((ext_vector_type(8))) float v8f;

#if defined(__has_builtin)
#  if __has_builtin(__builtin_amdgcn_wmma_f32_16x16x4_f32)
#    define HAVE_WMMA_F32X4 1
#  endif
#endif
#ifndef HAVE_WMMA_F32X4
#  define HAVE_WMMA_F32X4 0
#endif

// D(16x16,f32) = A(16x4,f32) * B(4x16,f32) + C ; CDNA5 wave32 fragment layouts:
//   A: lane m=lane&15, khalf=lane>>4, a[j] = A[m][2*khalf+j]
//   B: lane n=lane&15, khalf=lane>>4, b[j] = B[2*khalf+j][n]
//   C: lane n=lane&15, mh=lane>>4,    c[r] = C[r+8*mh][n]
__device__ inline v8f wmma16x16x4(v2f a, v2f b, v8f c) {
#if HAVE_WMMA_F32X4
  return __builtin_amdgcn_wmma_f32_16x16x4_f32(false, a, false, b, (short)0, c,
                                               false, false);
#else
  // layout-exact shuffle fallback (compiles everywhere; histogram reveals no-wmma)
  const int lane = threadIdx.x & 31;
  const int nl = lane & 15;
  const int mh = lane >> 4;
  #pragma unroll
  for (int r = 0; r < 8; ++r) {
    const int M = r + 8 * mh;
    #pragma unroll
    for (int kk = 0; kk < 4; ++kk) {
      float av = __shfl(kk & 1 ? a.y : a.x, M  + ((kk & 2) << 3), 32);
      float bv = __shfl(kk & 1 ? b.y : b.x, nl + ((kk & 2) << 3), 32);
      c[r] += av * bv;
    }
  }
  return c;
#endif
}

// ------- 256x256 transpose into pair-interleaved B-fragment layout ----------
// Wp[(k>>1)*512 + 2n + (k&1)] = W[n][k]  ->  a B-fragment (b[0],b[1]) is one
// contiguous 8B v2f, and 16 consecutive lanes cover 128B (fully coalesced).
__global__ __launch_bounds__(256)
void transpose3_kernel(const float* __restrict__ Wq, const float* __restrict__ Wk,
                       const float* __restrict__ Wv, float* __restrict__ WqP,
                       float* __restrict__ WkP, float* __restrict__ WvP)
{
  __shared__ float tile[32][33];
  const float* src; float* dst;
  if (blockIdx.z == 0)      { src = Wq; dst = WqP; }
  else if (blockIdx.z == 1) { src = Wk; dst = WkP; }
  else                      { src = Wv; dst = WvP; }
  const int tx = threadIdx.x & 31;
  const int ty = threadIdx.x >> 5;           // 0..7
  const int bx = blockIdx.x * 32, by = blockIdx.y * 32;
  #pragma unroll
  for (int i = 0; i < 32; i += 8)
    tile[ty + i][tx] = src[(size_t)(by + ty + i) * F + bx + tx];
  __syncthreads();
  #pragma unroll
  for (int i = 0; i < 32; i += 8) {
    const int k = bx + ty + i;               // column of W
    const int n = by + tx;                   // row of W
    dst[(size_t)(k >> 1) * 512 + 2 * n + (k & 1)] = tile[tx][ty + i];
  }
}

// ---------------- QKV projection: WMMA f32 16x16x4 --------------------------
// Block = 128 threads (4 waves). Block owns a 16-row strip of x, copied into
// LDS with global_load_async_to_lds_b128 (ASYNCcnt path, padded pitch 260
// floats -> conflict-free ds_load_b64 A-frags). Each wave owns 64 output
// columns (4 accumulator tiles); k steps of 4; all three weight matrices are
// processed against the same LDS tile.
#define XPITCH 260

__global__ __launch_bounds__(128)
void qkv_gemm_kernel(const float* __restrict__ x,
                     const float* __restrict__ WqP, const float* __restrict__ WkP,
                     const float* __restrict__ WvP,
                     float* __restrict__ qb, float* __restrict__ kb,
                     float* __restrict__ vb)
{
  __shared__ float xs[16 * XPITCH];
  const int tid  = threadIdx.x;
  const int lane = tid & 31;
  const int wid  = tid >> 5;
  const size_t m0 = (size_t)blockIdx.x * 16;

  { // async copy 16x256 floats (1024 float4) straight into LDS
    const float* src = x + m0 * F;
    #pragma unroll
    for (int i = 0; i < 8; ++i) {
      const int e   = tid + i * 128;                 // float4 index
      const int row = e >> 6;
      const int col = (e & 63) << 2;
      const unsigned lds_addr =
          (unsigned)(uintptr_t)(const void*)&xs[row * XPITCH + col];
      const unsigned long long gaddr =
          (unsigned long long)(uintptr_t)(const void*)(src + e * 4);
      asm volatile("global_load_async_to_lds_b128 %0, %1, off"
                   :: "v"(lds_addr), "v"(gaddr) : "memory");
    }
    asm volatile("s_wait_asynccnt 0x0" ::: "memory");
  }
  __syncthreads();

  const int nl = lane & 15;       // N (and A's M) sub-index
  const int kh = lane >> 4;       // K half for A/B frags, M half for C
  const int n0 = wid * 64;

  const float* Ws[3] = {WqP, WkP, WvP};
  float*       Os[3] = {qb, kb, vb};

  for (int w = 0; w < 3; ++w) {
    const float* Wp = Ws[w];
    __builtin_prefetch(Wp + (size_t)kh * 512 + 2 * (n0 + nl), 0, 0);
    v8f acc0 = {}, acc1 = {}, acc2 = {}, acc3 = {};
    for (int k0 = 0; k0 < F; k0 += 4) {
      const v2f a = *(const v2f*)&xs[nl * XPITCH + k0 + 2 * kh];
      const float* bp = Wp + ((size_t)(k0 >> 1) + kh) * 512 + 2 * (n0 + nl);
      const v2f b0 = *(const v2f*)(bp + 0);    // {W[n][k], W[n][k+1]}
      const v2f b1 = *(const v2f*)(bp + 32);
      const v2f b2 = *(const v2f*)(bp + 64);
      const v2f b3 = *(const v2f*)(bp + 96);
      acc0 = wmma16x16x4(a, b0, acc0);
      acc1 = wmma16x16x4(a, b1, acc1);
      acc2 = wmma16x16x4(a, b2, acc2);
      acc3 = wmma16x16x4(a, b3, acc3);
    }
    float* o = Os[w];
    #pragma unroll
    for (int r = 0; r < 8; ++r) {
      const size_t orow = (m0 + r + 8 * kh) * F + n0 + nl;
      o[orow]      = acc0[r];
      o[orow + 16] = acc1[r];
      o[orow + 32] = acc2[r];
      o[orow + 48] = acc3[r];
    }
  }
}

// ---------------- kNN: LDS-resident coords, 32-pass lexicographic min -------
// Block = 256 thr (8 waves), 64 queries/block, 1 query per wave at a time.
// key = (bits(d2) << 13) | (j+1)  -> exact (d2, index) ordering like top_k.
__global__ __launch_bounds__(256)
void knn_kernel(const float* __restrict__ coords, int* __restrict__ nbr)
{
  __shared__ float xs[S], ys[S], zs[S];          // 48 KB
  const int tid = threadIdx.x, lane = tid & 31, wid = tid >> 5;
  const int b  = blockIdx.x >> 6;
  const int q0 = (blockIdx.x & 63) * 64;
  const float* cb = coords + (size_t)b * S * 3;

  for (int p = tid; p < S; p += 256) {
    xs[p] = cb[3 * p + 0];
    ys[p] = cb[3 * p + 1];
    zs[p] = cb[3 * p + 2];
  }
  __syncthreads();

  for (int qi = 0; qi < 8; ++qi) {
    const int s = q0 + wid * 8 + qi;
    const float qx = xs[s], qy = ys[s], qz = zs[s];
    unsigned long long prev = 0;
    int* orow = nbr + ((size_t)b * S + s) * NK;
    for (int p = 0; p < NK; ++p) {
      unsigned long long best = ~0ull;
      for (int j = lane; j < S; j += 32) {
        const float dx = xs[j] - qx, dy = ys[j] - qy, dz = zs[j] - qz;
        const float d2 = fmaf(dx, dx, fmaf(dy, dy, dz * dz));
        const unsigned int bits = (j == s) ? 0xFFFFFFFFu : __float_as_uint(d2);
        const unsigned long long key =
            ((unsigned long long)bits << 13) | (unsigned)(j + 1);
        if (key > prev && key < best) best = key;
      }
      #pragma unroll
      for (int off = 16; off > 0; off >>= 1) {
        const unsigned long long o = __shfl_xor(best, off, 32);
        best = (o < best) ? o : best;
      }
      prev = best;
      if (lane == 0) orow[p] = (int)(best & 0x1FFFu) - 1;
    }
  }
}

// ---------------- gathered neighbor attention: one wave per (b,h,s) ---------
__global__ __launch_bounds__(256)
void attn_kernel(const float* __restrict__ qb, const float* __restrict__ kb,
                 const float* __restrict__ vb, const int* __restrict__ nbr,
                 float* __restrict__ out)
{
  const int lane = threadIdx.x & 31;
  const int wid  = blockIdx.x * 8 + (threadIdx.x >> 5);   // 0 .. B*H*S-1
  const int s = wid & (S - 1);
  const int h = (wid >> 12) & (H - 1);
  const int b = wid >> 15;
  const size_t row = (size_t)b * S + s;

  const int   j  = nbr[row * NK + lane];                  // lane = neighbor k
  const float qv = qb[row * F + h * D + lane];            // lane = dim d

  // score_k: lane k owns its neighbor's key row (32 floats)
  float kr[32];
  const size_t krow = ((size_t)b * S + j) * F + h * D;
  #pragma unroll
  for (int d4 = 0; d4 < 8; ++d4) {
    const float4 t = *(const float4*)(kb + krow + d4 * 4);
    kr[d4 * 4 + 0] = t.x; kr[d4 * 4 + 1] = t.y;
    kr[d4 * 4 + 2] = t.z; kr[d4 * 4 + 3] = t.w;
  }
  float sc = 0.f;
  #pragma unroll
  for (int d = 0; d < 32; ++d) sc += __shfl(qv, d, 32) * kr[d];
  sc *= 0.17677669529663688f;                             // 32^-0.5

  // softmax across the 32 lanes (neighbors)
  float m = sc;
  #pragma unroll
  for (int off = 16; off > 0; off >>= 1) m = fmaxf(m, __shfl_xor(m, off, 32));
  const float e = __expf(sc - m);
  float sum = e;
  #pragma unroll
  for (int off = 16; off > 0; off >>= 1) sum += __shfl_xor(sum, off, 32);
  const float p = e / sum;

  // out_d = sum_k p_k * V[j_k][d] ; lane = d, V loads coalesced (128B/iter)
  float acc = 0.f;
  #pragma unroll
  for (int k = 0; k < 32; ++k) {
    const float pk = __shfl(p, k, 32);
    const int   jk = __shfl(j, k, 32);
    acc += pk * vb[((size_t)b * S + jk) * F + h * D + lane];
  }
  out[row * F + h * D + lane] = acc;
}

// ---------------- metric = mean over heads of K -----------------------------
__global__ __launch_bounds__(256)
void metric_kernel(const float* __restrict__ kb, float* __restrict__ mout)
{
  const int t = blockIdx.x * 256 + threadIdx.x;           // over B*S*D
  if (t >= B * S * D) return;
  const int dj = t & (D - 1);
  const size_t row = (size_t)(t >> 5);
  float s = 0.f;
  #pragma unroll
  for (int h = 0; h < H; ++h) s += kb[row * F + h * D + dj];
  mout[t] = s * (1.0f / H);
}

// ---------------- launcher ---------------------------------------------------
extern "C" void kernel_launch(void* const* d_in, const int* in_sizes, int n_in,
                              void* d_out, int out_size, void* d_ws, size_t ws_size,
                              hipStream_t stream)
{
  (void)in_sizes; (void)n_in; (void)out_size; (void)ws_size;
  const float* x      = (const float*)d_in[0];
  const float* coords = (const float*)d_in[1];
  const float* Wq     = (const float*)d_in[2];
  const float* Wk     = (const float*)d_in[3];
  const float* Wv     = (const float*)d_in[4];

  float* out    = (float*)d_out;                      // [B,S,F]
  float* metric = out + (size_t)B * S * F;            // [B,S,D]

  char* ws = (char*)d_ws;
  size_t off = 0;
  auto alloc = [&](size_t bytes) -> void* {
    void* p = ws + off;
    off += (bytes + 255) & ~(size_t)255;
    return p;
  };
  float* qb  = (float*)alloc((size_t)B * S * F * 4);
  float* kb  = (float*)alloc((size_t)B * S * F * 4);
  float* vb  = (float*)alloc((size_t)B * S * F * 4);
  float* WqP = (float*)alloc((size_t)F * F * 4);
  float* WkP = (float*)alloc((size_t)F * F * 4);
  float* WvP = (float*)alloc((size_t)F * F * 4);
  int*   nbr = (int*)  alloc((size_t)B * S * NK * 4);

  transpose3_kernel<<<dim3(8, 8, 3), 256, 0, stream>>>(Wq, Wk, Wv, WqP, WkP, WvP);
  qkv_gemm_kernel<<<B * S / 16, 128, 0, stream>>>(x, WqP, WkP, WvP, qb, kb, vb);
  knn_kernel<<<B * (S / 64), 256, 0, stream>>>(coords, nbr);
  attn_kernel<<<(B * H * S) / 8, 256, 0, stream>>>(qb, kb, vb, nbr, out);
  metric_kernel<<<(B * S * D) / 256, 256, 0, stream>>>(kb, metric);
}